// FusionNet_47510928228768
// MI455X (gfx1250) — compile-verified
//
#include <hip/hip_runtime.h>
#include <math.h>

typedef __attribute__((ext_vector_type(2))) float v2f;
typedef __attribute__((ext_vector_type(8))) float v8f;

#define Bn 8
#define Cc 32
#define Hh 144
#define Ww 144
#define Ll 2304   // 48*48
#define Ff 288    // C*9
#define HW (Hh*Ww)

// ---------------------------------------------------------------------------
// Kernel 1: unfold3 + L2-normalize, one wave per patch row. Output (B, L, F).
// ---------------------------------------------------------------------------
__global__ void prep_rows_kernel(const float* __restrict__ x, float* __restrict__ out) {
  int wid  = blockIdx.x * 8 + (threadIdx.x >> 5);   // 2304 blocks * 8 waves = 18432 rows
  int lane = threadIdx.x & 31;
  int b = wid / Ll;
  int l = wid % Ll;
  int lh = l / 48, lw = l % 48;

  float vals[9];
  float ss = 0.f;
#pragma unroll
  for (int k = 0; k < 9; ++k) {
    int f = lane + k * 32;              // F = 288 = 9*32 exactly
    int c = f / 9, r = f % 9;
    int i = r / 3, j = r % 3;
    int y  = lh * 3 + i - 1;
    int xx = lw * 3 + j - 1;
    float v = 0.f;
    if (y >= 0 && xx >= 0)
      v = x[(((size_t)b * Cc + c) * Hh + y) * Ww + xx];
    vals[k] = v;
    ss += v * v;
  }
  for (int off = 16; off > 0; off >>= 1) ss += __shfl_xor(ss, off, 32);
  float inv = 1.0f / fmaxf(sqrtf(ss), 1e-12f);

  size_t base = ((size_t)b * Ll + l) * Ff;
#pragma unroll
  for (int k = 0; k < 9; ++k) out[base + lane + k * 32] = vals[k] * inv;
}

// ---------------------------------------------------------------------------
// Kernel 2: sim GEMM (fp32 WMMA 16x16x4, 4-way split-K) fused with streaming
// per-row top-2. B tiles staged with async global->LDS; next tile prefetched.
// ---------------------------------------------------------------------------
__global__ __launch_bounds__(64) void sim_top2_kernel(const float* __restrict__ x8u,
                                                      const float* __restrict__ x4uT,
                                                      int* __restrict__ id2o) {
  __shared__ float4 ldsA4[2304];        // 32 rows x 288 floats (A strips, row-major)
  __shared__ float4 ldsB4[1152];        // 16 rows x 288 floats (B tile, [n][k])
  __shared__ float  ldsC[2][16][16];    // per-wave accumulator bounce

  float* ldsA = (float*)ldsA4;
  float* ldsB = (float*)ldsB4;

  int tid  = threadIdx.x;
  int wave = tid >> 5;
  int lane = tid & 31;
  int b  = blockIdx.x / 72;             // 576 workgroups = 8 * (144/2)
  int pr = blockIdx.x % 72;
  int l0 = pr * 32;                     // 32 consecutive sim rows per workgroup

  // Stage A: 32 consecutive rows of x8u -> contiguous 9216 floats.
  const float4* srcA = (const float4*)(x8u + ((size_t)b * Ll + l0) * Ff);
  for (int i = tid; i < 2304; i += 64) ldsA4[i] = srcA[i];

  int hi = lane >> 4;                   // half-wave select (K pair 0/1 vs 2/3)
  int lo = lane & 15;                   // M (for A) / N (for B)
  const float* aBase = ldsA + (wave * 16 + lo) * Ff + 2 * hi;
  const float* bBase = ldsB + lo * Ff + 2 * hi;

  float v1 = -3.4e38f, v2 = -3.4e38f;
  int   i1 = 0,        i2 = 0;

  for (int mt = 0; mt < 144; ++mt) {
    int m0 = mt * 16;
    __syncthreads();                    // prior K-loop done before restaging B

    // Async global->LDS staging of the 16x288 B tile (GV mode, B128/lane).
    {
      const float4* srcB = (const float4*)(x4uT + ((size_t)b * Ll + m0) * Ff);
      for (int i = tid; i < 1152; i += 64) {
        unsigned           dst = (unsigned)(uintptr_t)(const void*)(ldsB4 + i);
        unsigned long long src = (unsigned long long)(uintptr_t)(srcB + i);
        asm volatile("global_load_async_to_lds_b128 %0, %1, off"
                     :: "v"(dst), "v"(src) : "memory");
      }
      asm volatile("s_wait_asynccnt 0x0" ::: "memory");
    }
    __syncthreads();

    // Prefetch next B tile into cache while this tile computes.
    if (mt + 1 < 144) {
      const char* nb = (const char*)(x4uT + ((size_t)b * Ll + m0 + 16) * Ff);
      for (int i = tid; i < 144; i += 64) __builtin_prefetch(nb + (size_t)i * 128, 0, 0);
    }

    // 4-way split-K: independent accumulators break the WMMA RAW chain.
    v8f c0 = {}, c1 = {}, c2 = {}, c3 = {};
#pragma unroll 2
    for (int kt = 0; kt < 72; kt += 4) {  // K = 288 = 72 * 4
      v2f a0 = *(const v2f*)(aBase + (kt + 0) * 4);
      v2f b0 = *(const v2f*)(bBase + (kt + 0) * 4);
      v2f a1 = *(const v2f*)(aBase + (kt + 1) * 4);
      v2f b1 = *(const v2f*)(bBase + (kt + 1) * 4);
      v2f a2 = *(const v2f*)(aBase + (kt + 2) * 4);
      v2f b2 = *(const v2f*)(bBase + (kt + 2) * 4);
      v2f a3 = *(const v2f*)(aBase + (kt + 3) * 4);
      v2f b3 = *(const v2f*)(bBase + (kt + 3) * 4);
      c0 = __builtin_amdgcn_wmma_f32_16x16x4_f32(false, a0, false, b0, (short)0, c0, false, false);
      c1 = __builtin_amdgcn_wmma_f32_16x16x4_f32(false, a1, false, b1, (short)0, c1, false, false);
      c2 = __builtin_amdgcn_wmma_f32_16x16x4_f32(false, a2, false, b2, (short)0, c2, false, false);
      c3 = __builtin_amdgcn_wmma_f32_16x16x4_f32(false, a3, false, b3, (short)0, c3, false, false);
    }
    v8f c = (c0 + c1) + (c2 + c3);

    // C layout: VGPR r -> row (r + 8*hi), column lo
#pragma unroll
    for (int r = 0; r < 8; ++r) ldsC[wave][r + 8 * hi][lo] = c[r];
    __syncthreads();

    if (lane < 16) {
#pragma unroll
      for (int n = 0; n < 16; ++n) {
        float s = ldsC[wave][lane][n];
        int   m = m0 + n;
        bool gt1 = s > v1;
        bool gt2 = s > v2;
        float nv2 = gt1 ? v1 : (gt2 ? s : v2);
        int   ni2 = gt1 ? i1 : (gt2 ? m : i2);
        v2 = nv2;
        i2 = ni2;
        i1 = gt1 ? m : i1;
        v1 = gt1 ? s : v1;
      }
    }
  }

  if (lane < 16)
    id2o[(size_t)b * Ll + l0 + wave * 16 + lane] = i2;
}

// ---------------------------------------------------------------------------
// Kernel 3: gather 2nd-best patch column + fold3 + 1x1 conv (64->32) + PReLU.
// One thread per output pixel; weights cached in LDS.
// ---------------------------------------------------------------------------
__global__ void conv1_gather_kernel(const float* __restrict__ x4,
                                    const float* __restrict__ x4uT,
                                    const int* __restrict__ id2b,
                                    const float* __restrict__ w1,
                                    const float* __restrict__ b1,
                                    const float* __restrict__ pa,
                                    float* __restrict__ hout) {
  __shared__ float ws[2048];            // w1: (32 out, 64 in)
  int tid = threadIdx.x;
  for (int i = tid; i < 2048; i += 256) ws[i] = w1[i];
  __syncthreads();

  int t = blockIdx.x * 256 + tid;       // 648 * 256 = 8*144*144 exactly
  int b   = t / HW;
  int rem = t % HW;
  int y = rem / Ww;
  int x = rem % Ww;

  float xin[32], tin[32];
#pragma unroll
  for (int c = 0; c < 32; ++c)
    xin[c] = x4[(((size_t)b * Cc + c) * Hh + y) * Ww + x];

  if (y == Hh - 1 || x == Ww - 1) {
#pragma unroll
    for (int c = 0; c < 32; ++c) tin[c] = 0.f;
  } else {
    int Y = y + 1, X = x + 1;
    int lh = Y / 3, i = Y % 3, lw = X / 3, j = X % 3;
    int l = lh * 48 + lw;
    int m = id2b[(size_t)b * Ll + l];
    const float* src = x4uT + ((size_t)b * Ll + m) * Ff + i * 3 + j;
#pragma unroll
    for (int c = 0; c < 32; ++c) tin[c] = src[c * 9];
  }

  float a = pa[0];
  for (int co = 0; co < 32; ++co) {
    float acc = b1[co];
    const float* wr = ws + co * 64;
#pragma unroll
    for (int c = 0; c < 32; ++c) acc = fmaf(wr[c], xin[c], acc);
#pragma unroll
    for (int c = 0; c < 32; ++c) acc = fmaf(wr[32 + c], tin[c], acc);
    acc = (acc >= 0.f) ? acc : a * acc;
    hout[(((size_t)b * Cc + co) * Hh + y) * Ww + x] = acc;
  }
}

// ---------------------------------------------------------------------------
// Kernel 4: 3x3 conv (32->32, pad 1). Block = 16x16 pixel tile, halo in LDS.
// ---------------------------------------------------------------------------
__global__ void conv2_kernel(const float* __restrict__ hin,
                             const float* __restrict__ w2,
                             const float* __restrict__ b2,
                             float* __restrict__ out) {
  __shared__ float hs[32][18][18];
  int tid = threadIdx.x;
  int bi = blockIdx.x;                  // 648 = 8 * 9 * 9
  int b  = bi / 81;
  int tt = bi % 81;
  int y0 = (tt / 9) * 16;
  int x0 = (tt % 9) * 16;

  for (int idx = tid; idx < 32 * 18 * 18; idx += 256) {
    int ci = idx / 324;
    int r  = idx % 324;
    int yy = r / 18, xx = r % 18;
    int gy = y0 + yy - 1, gx = x0 + xx - 1;
    float v = 0.f;
    if (gy >= 0 && gy < Hh && gx >= 0 && gx < Ww)
      v = hin[(((size_t)b * Cc + ci) * Hh + gy) * Ww + gx];
    hs[ci][yy][xx] = v;
  }
  __syncthreads();

  int ty = tid / 16, tx = tid % 16;
  for (int co = 0; co < 32; ++co) {
    float acc = b2[co];
    for (int ci = 0; ci < 32; ++ci) {
      const float* w = w2 + ((size_t)co * 32 + ci) * 9;
#pragma unroll
      for (int dy = 0; dy < 3; ++dy)
#pragma unroll
        for (int dx = 0; dx < 3; ++dx)
          acc = fmaf(w[dy * 3 + dx], hs[ci][ty + dy][tx + dx], acc);
    }
    out[(((size_t)b * Cc + co) * Hh + y0 + ty) * Ww + x0 + tx] = acc;
  }
}

// ---------------------------------------------------------------------------
extern "C" void kernel_launch(void* const* d_in, const int* in_sizes, int n_in,
                              void* d_out, int out_size, void* d_ws, size_t ws_size,
                              hipStream_t stream) {
  (void)in_sizes; (void)n_in; (void)out_size; (void)ws_size;
  const float* x8   = (const float*)d_in[0];
  const float* x4   = (const float*)d_in[1];
  const float* w1   = (const float*)d_in[2];
  const float* b1   = (const float*)d_in[3];
  const float* pa   = (const float*)d_in[4];
  const float* w2   = (const float*)d_in[5];
  const float* b2   = (const float*)d_in[6];

  const size_t rowsElems = (size_t)Bn * Ll * Ff;        // 5,308,416 floats
  float* x8u  = (float*)d_ws;
  float* x4uT = x8u + rowsElems;
  int*   idb  = (int*)(x4uT + rowsElems);
  float* hbuf = (float*)(idb + (size_t)Bn * Ll);

  // 1) unfold + normalize both inputs into (B, L, F) row-major
  prep_rows_kernel<<<2304, 256, 0, stream>>>(x8, x8u);
  prep_rows_kernel<<<2304, 256, 0, stream>>>(x4, x4uT);

  // 2) fp32 WMMA similarity GEMM fused with per-row top-2 -> 2nd-best index
  sim_top2_kernel<<<576, 64, 0, stream>>>(x8u, x4uT, idb);

  // 3) gather + fold + 1x1 conv + PReLU
  conv1_gather_kernel<<<648, 256, 0, stream>>>(x4, x4uT, idb, w1, b1, pa, hbuf);

  // 4) 3x3 conv, pad 1 -> output
  conv2_kernel<<<648, 256, 0, stream>>>(hbuf, w2, b2, (float*)d_out);
}